// InterpreterUnit_5858335392361
// MI455X (gfx1250) — compile-verified
//
#include <hip/hip_runtime.h>

// Elementwise interpreter-select over [N,128] rows. Pure HBM-streaming op:
// ~0.06 FLOP/byte => bandwidth-bound on MI455X (23.3 TB/s). Strategy:
//   - one wave32 per row, float4 (b128) per lane
//   - scalar (readfirstlane) flags => uniform s_cbranch, skip unneeded loads
//   - non-temporal loads/stores (CDNA5 TH_NT) since working set >> 192MB L2

typedef float v4f __attribute__((ext_vector_type(4)));

#define IW 131   // inputs row width (3 flags + 128 payload)
#define DW 128   // children / output row width

__global__ __launch_bounds__(256) void interp_select_kernel(
    const float* __restrict__ inputs,    // [N, 131]
    const float* __restrict__ children,  // [2, N, 128]
    const int*   __restrict__ arities,   // [N]
    float* __restrict__ out,             // [N, 128]
    int n)
{
    const int wave = (int)((blockIdx.x * blockDim.x + threadIdx.x) >> 5);
    const int lane = (int)(threadIdx.x & 31u);
    if (wave >= n) return;

    const size_t row    = (size_t)wave;
    const float* inrow  = inputs + row * (size_t)IW;
    const size_t base   = row * (size_t)DW + (size_t)(lane << 2);
    const v4f*   c0p    = (const v4f*)(children + base);
    const v4f*   c1p    = (const v4f*)(children + (size_t)n * (size_t)DW + base);

    // Flags are identical across the wave (same address every lane):
    // broadcast to SGPRs so the 4-way select becomes uniform scalar branches.
    const float f0 = inrow[0];
    const float f1 = inrow[1];
    const float f2 = inrow[2];
    const int cond_add = __builtin_amdgcn_readfirstlane(f0 > 0.0f ? 1 : 0);
    const int cond_mul = __builtin_amdgcn_readfirstlane(f1 > 0.0f ? 1 : 0);
    const int cond_sub = __builtin_amdgcn_readfirstlane(f2 > 0.0f ? 1 : 0);

    v4f r;
    if (cond_add) {
        const v4f a = __builtin_nontemporal_load(c0p);
        const v4f b = __builtin_nontemporal_load(c1p);
        r = a + b;
    } else if (cond_mul) {
        const v4f a = __builtin_nontemporal_load(c0p);
        const v4f b = __builtin_nontemporal_load(c1p);
        r = a * b;
    } else if (cond_sub) {
        const v4f a = __builtin_nontemporal_load(c0p);
        const int is_unary =
            __builtin_amdgcn_readfirstlane(arities[row] == 1 ? 1 : 0);
        if (is_unary) {
            r = -a;
        } else {
            const v4f b = __builtin_nontemporal_load(c1p);
            r = a - b;
        }
    } else {
        // Passthrough inputs[row, 3:131]. Row body starts at float offset
        // 131*row + 3 -> only 4-byte aligned, so use scalar loads.
        const float* fp = inrow + 3 + (lane << 2);
        r.x = fp[0];
        r.y = fp[1];
        r.z = fp[2];
        r.w = fp[3];
    }

    __builtin_nontemporal_store(r, (v4f*)(out + base));
}

extern "C" void kernel_launch(void* const* d_in, const int* in_sizes, int n_in,
                              void* d_out, int out_size, void* d_ws, size_t ws_size,
                              hipStream_t stream) {
    const float* inputs   = (const float*)d_in[0];
    const float* children = (const float*)d_in[1];
    const int*   arities  = (const int*)d_in[2];
    float*       out      = (float*)d_out;

    const int n = in_sizes[0] / IW;          // number of rows (N)
    const int waves_per_block = 256 / 32;    // 8 waves per 256-thread block
    const int blocks = (n + waves_per_block - 1) / waves_per_block;

    interp_select_kernel<<<blocks, 256, 0, stream>>>(inputs, children, arities,
                                                     out, n);
}